// PopPredict_49357764166209
// MI455X (gfx1250) — compile-verified
//
#include <hip/hip_runtime.h>

typedef __attribute__((ext_vector_type(2))) float v2f;
typedef __attribute__((ext_vector_type(4))) float v4f;
typedef __attribute__((ext_vector_type(8))) float v8f;

#define ALPHA 0.3f
#define OMALPHA 0.7f

// Layout: one 16-row tile per wave-pair (K split across the pair).
// Block = 256 threads = 8 waves = 4 tiles. Grid = B / 64.
__global__ __launch_bounds__(256) void pop_predict_kernel(
    const int*   __restrict__ item,
    const int*   __restrict__ time_release,
    const int*   __restrict__ item_genre,     // (B,8)
    const int*   __restrict__ item_director,
    const int*   __restrict__ item_actor,     // (B,16)
    const int*   __restrict__ time_q,
    const float* __restrict__ pop_history,    // (B,256)
    const int*   __restrict__ valid_pop_len,
    const float* __restrict__ item_w,         // (NUM_ITEM,128)
    const float* __restrict__ time_w,         // (2049,128)
    const float* __restrict__ genre_w,        // (64,128)
    const float* __restrict__ dir_w,          // (10000,128)
    const float* __restrict__ actor_w,        // (100000,128)
    const float* __restrict__ W_time,         // (512)
    const float* __restrict__ b_time,         // (1)
    const float* __restrict__ W_side,         // (384)
    const float* __restrict__ b_side,         // (1)
    const float* __restrict__ attn_w,         // (3)
    float*       __restrict__ out,            // [pop|time|side|final], each B
    int B)
{
  // partial pre-activation dot products, per (kphase, tile-in-block, row)
  __shared__ float lds_t[2][4][16];
  __shared__ float lds_s[2][4][16];

  const int l    = threadIdx.x & 31;   // lane in wave32
  const int wv   = threadIdx.x >> 5;   // wave in block (0..7)
  const int kph  = wv & 1;             // K-split phase (cols 0..63 / 64..127)
  const int tib  = wv >> 1;            // tile in block (0..3)
  const int m    = l & 15;             // row within tile (WMMA M)
  const int h    = l >> 4;             // lane half -> K sub-pair
  const int tile = blockIdx.x * 4 + tib;
  const int base = tile * 16;
  const int row  = base + m;

  // Per-row gather indices (lanes l and l+16 duplicate the same row's indices).
  const int it_i = item[row];
  const int tr_i = time_release[row];
  const int tq_i = time_q[row];
  const int dr_i = item_director[row];

  // This lane's starting column inside the 128-wide embedding rows.
  const int cbase = kph * 64 + h * 2;

  // Hoisted, loop-invariant base pointers: every inner-loop load becomes a
  // global_load_b64 with a compile-time immediate offset (stride 16B x 16 steps).
  const float* pt  = time_w + (size_t)tq_i * 128 + cbase;
  const float* ptr = time_w + (size_t)tr_i * 128 + cbase;
  const float* pit = item_w + (size_t)it_i * 128 + cbase;
  const float* pd  = dir_w  + (size_t)dr_i * 128 + cbase;
  const float* pg[8];
#pragma unroll
  for (int g = 0; g < 8; ++g)
    pg[g] = genre_w + (size_t)item_genre[row * 8 + g] * 128 + cbase;
  const float* pa[16];
#pragma unroll
  for (int a = 0; a < 16; ++a)
    pa[a] = actor_w + (size_t)item_actor[row * 16 + a] * 128 + cbase;
  const float* pwt = W_time + cbase;   // +0/+128/+256/+384 segments via imm offsets
  const float* pws = W_side + cbase;   // +0/+128/+256 segments via imm offsets

  // B-matrix carries W only in column N==0 (lanes 0 and 16), zero elsewhere.
  const float sel = (m == 0) ? 1.0f : 0.0f;

  v8f c_time = {0.f, 0.f, 0.f, 0.f, 0.f, 0.f, 0.f, 0.f};
  v8f c_side = {0.f, 0.f, 0.f, 0.f, 0.f, 0.f, 0.f, 0.f};

#pragma unroll
  for (int kk = 0; kk < 16; ++kk) {
    const int o = kk * 4;   // float offset; 16B per step -> immediate offsets

    // --- gather A fragments straight from global ---
    v2f vt   = *(const v2f*)(pt + o);
    v2f vtr  = *(const v2f*)(ptr + o);
    v2f a_td = vtr - vt;                                    // temporal_dis
    v2f a_it = *(const v2f*)(pit + o);

    v2f gm = {0.f, 0.f};
#pragma unroll
    for (int g = 0; g < 8; ++g)
      gm += *(const v2f*)(pg[g] + o);
    gm *= 0.125f;                                           // genre mean

    v2f dv = *(const v2f*)(pd + o);

    v2f am = {0.f, 0.f};
#pragma unroll
    for (int a = 0; a < 16; ++a)
      am += *(const v2f*)(pa[a] + o);
    am *= 0.0625f;                                          // actor mean

    // --- B fragments: W segments broadcast into column 0 ---
    v2f wt0 = sel * (*(const v2f*)(pwt + o));               // vs temporal_dis
    v2f wt1 = sel * (*(const v2f*)(pwt + 128 + o));         // vs item
    v2f wt2 = sel * (*(const v2f*)(pwt + 256 + o));         // vs t
    v2f wt3 = sel * (*(const v2f*)(pwt + 384 + o));         // vs tr
    v2f ws0 = sel * (*(const v2f*)(pws + o));               // vs genre mean
    v2f ws1 = sel * (*(const v2f*)(pws + 128 + o));         // vs director
    v2f ws2 = sel * (*(const v2f*)(pws + 256 + o));         // vs actor mean

    // --- full-fp32 WMMA accumulation: D = A x B + C ---
    c_time = __builtin_amdgcn_wmma_f32_16x16x4_f32(false, a_td, false, wt0, (short)0, c_time, false, false);
    c_time = __builtin_amdgcn_wmma_f32_16x16x4_f32(false, a_it, false, wt1, (short)0, c_time, false, false);
    c_time = __builtin_amdgcn_wmma_f32_16x16x4_f32(false, vt,   false, wt2, (short)0, c_time, false, false);
    c_time = __builtin_amdgcn_wmma_f32_16x16x4_f32(false, vtr,  false, wt3, (short)0, c_time, false, false);

    c_side = __builtin_amdgcn_wmma_f32_16x16x4_f32(false, gm,   false, ws0, (short)0, c_side, false, false);
    c_side = __builtin_amdgcn_wmma_f32_16x16x4_f32(false, dv,   false, ws1, (short)0, c_side, false, false);
    c_side = __builtin_amdgcn_wmma_f32_16x16x4_f32(false, am,   false, ws2, (short)0, c_side, false, false);
  }

  // Column N==0 of D lives in lane 0 (rows 0..7 in VGPRs 0..7) and lane 16 (rows 8..15).
  if (m == 0) {
#pragma unroll
    for (int v = 0; v < 8; ++v) {
      const int r = h * 8 + v;
      lds_t[kph][tib][r] = c_time[v];
      lds_s[kph][tib][r] = c_side[v];
    }
  }
  __syncthreads();

  // Even wave of each pair finishes the tile: EMA scan + bias/relu + attention blend.
  if (kph == 0 && l < 16) {
    // EMA recurrence, exact replica of the reference scan, gathered at len-1.
    // b128 chunk loads; the serial FMA chain is the recurrence itself.
    const float* ph = pop_history + (size_t)row * 256;
    const int n = valid_pop_len[row] - 1;
    float ema = 0.0f;
#pragma unroll 4
    for (int c = 0; c < 64; ++c) {
      const v4f xv = *(const v4f*)(ph + c * 4);
#pragma unroll
      for (int e = 0; e < 4; ++e) {
        const int j = c * 4 + e;
        if (j == 0) {
          ema = xv[e];
        } else {
          const float cand = fmaf(ALPHA, xv[e], OMALPHA * ema);
          ema = (j <= n) ? cand : ema;
        }
      }
    }

    const float bt = b_time[0];
    const float bs = b_side[0];
    const float tv = fmaxf(lds_t[0][tib][l] + lds_t[1][tib][l] + bt, 0.0f);
    const float sv = fmaxf(lds_s[0][tib][l] + lds_s[1][tib][l] + bs, 0.0f);

    // softmax over the 3 attention logits
    const float a0 = attn_w[0], a1 = attn_w[1], a2 = attn_w[2];
    const float mx = fmaxf(a0, fmaxf(a1, a2));
    const float e0 = __expf(a0 - mx), e1 = __expf(a1 - mx), e2 = __expf(a2 - mx);
    const float inv = 1.0f / (e0 + e1 + e2);
    const float fin = (e0 * ema + e1 * tv + e2 * sv) * inv;

    out[row]         = ema;   // pop_history_output
    out[B + row]     = tv;    // time_output
    out[2 * B + row] = sv;    // sideinfo_output
    out[3 * B + row] = fin;   // output
  }
}

extern "C" void kernel_launch(void* const* d_in, const int* in_sizes, int n_in,
                              void* d_out, int out_size, void* d_ws, size_t ws_size,
                              hipStream_t stream) {
  const int B = in_sizes[0];        // 8192
  const int blocks = B / 64;        // 4 tiles (64 rows) per 256-thread block
  pop_predict_kernel<<<blocks, 256, 0, stream>>>(
      (const int*)d_in[0],    // item
      (const int*)d_in[1],    // time_release
      (const int*)d_in[2],    // item_genre
      (const int*)d_in[3],    // item_director
      (const int*)d_in[4],    // item_actor
      (const int*)d_in[5],    // time
      (const float*)d_in[6],  // pop_history
      (const int*)d_in[8],    // valid_pop_len (d_in[7]=pop_gt unused)
      (const float*)d_in[9],  // embed_item_w
      (const float*)d_in[10], // embed_time_w
      (const float*)d_in[11], // embed_genre_w
      (const float*)d_in[12], // embed_director_w
      (const float*)d_in[13], // embed_actor_w
      (const float*)d_in[14], // W_time
      (const float*)d_in[15], // b_time
      (const float*)d_in[16], // W_side
      (const float*)d_in[17], // b_side
      (const float*)d_in[18], // attn_w
      (float*)d_out, B);
}